// MultiHeadGraphChannelAttention_15427522527277
// MI455X (gfx1250) — compile-verified
//
#include <hip/hip_runtime.h>
#include <hip/hip_bf16.h>
#include <math.h>

typedef float v2f __attribute__((ext_vector_type(2)));
typedef float v8f __attribute__((ext_vector_type(8)));

#define NN    4096      // batch of gathered nodes
#define NODES 8192
#define INF_  256       // in features
#define OUTF  128       // out features per head
#define NHEAD 4
#define HKPAD 136       // floats per LDS row: 16B-aligned, +16 bank shift per 2 rows

// ---------------------------------------------------------------------------
// Kernel 1: adj_b bit-pack.  bits[i][t] bit b = (adj[idx[i]][idx[t*32+b]] > 0)
// grid: NN blocks x 128 threads
// ---------------------------------------------------------------------------
__global__ void k_pack_adj(const int* __restrict__ adj,
                           const int* __restrict__ idx,
                           unsigned* __restrict__ bits) {
    const int i = blockIdx.x;
    const int t = threadIdx.x;                 // 0..127
    const long ri = idx[i];
    const int* __restrict__ row = adj + ri * (long)NODES;
    const int j0 = t * 32;
    unsigned w = 0;
#pragma unroll 8
    for (int b = 0; b < 32; ++b) {
        const int rj = idx[j0 + b];
        w |= (row[rj] > 0 ? 1u : 0u) << b;
    }
    bits[(size_t)i * 128 + t] = w;
}

// ---------------------------------------------------------------------------
// Kernel 2: h[head][n][o] = sum_f x[idx[n]][f] * W[head][f][o]
// grid: (NN/16, NHEAD) x 256 threads (8 waves, one 16x16 col tile each)
// Uses V_WMMA_F32_16X16X4_F32.
// ---------------------------------------------------------------------------
__global__ void k_gemm_h(const float* __restrict__ x,
                         const int*   __restrict__ idx,
                         const float* __restrict__ W,
                         float* __restrict__ h_ws) {
    const int lane = threadIdx.x & 31;
    const int wave = threadIdx.x >> 5;         // 0..7 -> column tile
    const int head = blockIdx.y;
    const int m0   = blockIdx.x * 16;
    const int n0   = wave * 16;

    const int mloc = lane & 15;
    const int half = lane >> 4;                // 0/1 -> K sub-slot

    const long rowA = idx[m0 + mloc];
    const float* __restrict__ xrow = x + rowA * INF_;
    const float* __restrict__ wcol = W + ((size_t)head * INF_) * OUTF + (n0 + mloc);

    v8f acc = {};
#pragma unroll 8
    for (int k = 0; k < INF_; k += 4) {
        const int kA = k + 2 * half;
        v2f a = *(const v2f*)(xrow + kA);                          // K=kA, kA+1
        v2f b;
        b.x = wcol[(size_t)kA * OUTF];
        b.y = wcol[(size_t)(kA + 1) * OUTF];
        acc = __builtin_amdgcn_wmma_f32_16x16x4_f32(
                  false, a, false, b, (short)0, acc, false, false);
    }

    float* __restrict__ hd = h_ws + (((size_t)head * NN + m0) * OUTF) + n0 + mloc;
#pragma unroll
    for (int r = 0; r < 8; ++r) {
        hd[(size_t)(r + 8 * half) * OUTF] = acc[r];                // M = r + 8*half
    }
}

// ---------------------------------------------------------------------------
// Kernel 3: s1[h][n] = h[h][n] . a1[h] ; s2 likewise with a2
// ---------------------------------------------------------------------------
__global__ void k_scores(const float* __restrict__ h_ws,
                         const float* __restrict__ a,
                         float* __restrict__ s1,
                         float* __restrict__ s2) {
    const int tid  = blockIdx.x * blockDim.x + threadIdx.x;        // 0..16383
    const int head = tid >> 12;
    const float* __restrict__ hr = h_ws + (size_t)tid * OUTF;
    const float* __restrict__ a1 = a + (size_t)head * (2 * OUTF);
    const float* __restrict__ a2 = a1 + OUTF;
    float sA = 0.f, sB = 0.f;
#pragma unroll 8
    for (int o = 0; o < OUTF; ++o) {
        const float v = hr[o];
        sA += v * a1[o];
        sB += v * a2[o];
    }
    s1[tid] = sA;
    s2[tid] = sB;
}

__device__ inline float wave_max32(float v) {
#pragma unroll
    for (int off = 16; off > 0; off >>= 1) v = fmaxf(v, __shfl_xor(v, off, 32));
    return v;
}
__device__ inline float wave_sum32(float v) {
#pragma unroll
    for (int off = 16; off > 0; off >>= 1) v += __shfl_xor(v, off, 32);
    return v;
}

// ---------------------------------------------------------------------------
// Kernel 4: per (head,row) masked leaky-relu softmax statistics.
// grid: (NHEAD*NN/4) x 128 threads (4 waves, one row each)
// ---------------------------------------------------------------------------
__global__ void k_stats(const float* __restrict__ s1,
                        const float* __restrict__ s2,
                        const unsigned* __restrict__ bits,
                        float* __restrict__ rmax,
                        float* __restrict__ rsum) {
    const int gidx = blockIdx.x * 4 + (threadIdx.x >> 5);          // head*NN + i
    const int head = gidx >> 12;
    const int i    = gidx & (NN - 1);
    const int lane = threadIdx.x & 31;

    const float s1v = s1[gidx];
    const unsigned* __restrict__ brow = bits + (size_t)i * 128;
    const float* __restrict__ s2h = s2 + (size_t)head * NN;

    float mmax = -3.0e38f;
    for (int it = 0; it < 128; ++it) {
        const unsigned w = brow[it];                               // uniform
        if (!w) continue;
        float e = s1v + s2h[it * 32 + lane];
        e = (e < 0.f) ? 0.2f * e : e;
        if ((w >> lane) & 1u) mmax = fmaxf(mmax, e);
    }
    mmax = wave_max32(mmax);

    float ssum = 0.f;
    if (mmax > -2.0e38f) {
        for (int it = 0; it < 128; ++it) {
            const unsigned w = brow[it];
            if (!w) continue;
            float e = s1v + s2h[it * 32 + lane];
            e = (e < 0.f) ? 0.2f * e : e;
            if ((w >> lane) & 1u) ssum += __expf(e - mmax);
        }
        ssum = wave_sum32(ssum);
    }
    if (lane == 0) { rmax[gidx] = mmax; rsum[gidx] = ssum; }
}

// ---------------------------------------------------------------------------
// Kernel 5: fused  out[n][head*128+o] = elu( sum_j P[i][j] * h[head][j][o] )
// Key tiles DMA'd into a double-buffered LDS via GLOBAL_LOAD_ASYNC_TO_LDS_B128
// (tracked with ASYNCcnt) so the tile-(k+1) load overlaps the tile-k WMMAs.
// s2 row cached in LDS.  WMMA f32 16x16x4, 4 col tiles per wave.
// grid: (NN/64, NHEAD) x 256 threads (8 waves)
// ---------------------------------------------------------------------------
__global__ void k_attn(const float* __restrict__ h_ws,
                       const float* __restrict__ s1,
                       const float* __restrict__ s2,
                       const unsigned* __restrict__ bits,
                       const float* __restrict__ rmax,
                       const float* __restrict__ rsum,
                       float* __restrict__ out) {
    __shared__ __align__(16) float hk[2][16][HKPAD];   // 2 x 8.5 KB key tiles
    __shared__ float s2s[NN];                          // 16 KB score cache

    const int lane  = threadIdx.x & 31;
    const int qt    = threadIdx.x >> 6;            // 0..3 query tile in block
    const int chalf = (threadIdx.x >> 5) & 1;      // 0/1 -> col tiles 0-3 / 4-7
    const int head  = blockIdx.y;
    const int m0    = blockIdx.x * 64 + qt * 16;

    const int mloc = lane & 15;
    const int half = lane >> 4;

    const int gi   = m0 + mloc;
    const int sidx = head * NN + gi;
    const float s1v  = s1[sidx];
    const float rmx  = rmax[sidx];
    const float rsm  = rsum[sidx];
    const bool  uni  = (rsm <= 0.f);               // isolated row -> uniform softmax
    const float rinv = uni ? 0.f : 1.f / rsm;
    const unsigned* __restrict__ brow = bits + (size_t)gi * 128;
    const float* __restrict__ hh = h_ws + (size_t)head * NN * OUTF;

    // ---- one-time: cache this head's s2 row in LDS (16 KB) ----
    {
        const float* __restrict__ s2h = s2 + (size_t)head * NN;
        for (int t = threadIdx.x * 4; t < NN; t += 256 * 4)
            *(float4*)&s2s[t] = *(const float4*)&s2h[t];
    }

    // per-thread slot for cooperative tile DMA: 2 x b128 per thread per tile
    const int ldr = threadIdx.x >> 4;              // 0..15 key row
    const int ldc = (threadIdx.x & 15) * 8;        // 0..120 col base

    // issue async DMA of key tile kt into buffer buf
    auto issue_tile = [&](int kt, int buf) {
        const float* src = hh + (size_t)(kt * 16 + ldr) * OUTF + ldc;
        unsigned l0 = (unsigned)(uintptr_t)&hk[buf][ldr][ldc];
        unsigned l1 = (unsigned)(uintptr_t)&hk[buf][ldr][ldc + 4];
        asm volatile("global_load_async_to_lds_b128 %0, %1, off"
                     :: "v"(l0), "v"(src) : "memory");
        asm volatile("global_load_async_to_lds_b128 %0, %1, off"
                     :: "v"(l1), "v"(src + 4) : "memory");
    };

    v8f acc[4] = {};

    issue_tile(0, 0);                              // prologue
    for (int kt = 0; kt < NN / 16; ++kt) {
        const int buf = kt & 1;
        if (kt + 1 < NN / 16) {
            issue_tile(kt + 1, buf ^ 1);           // overlap next tile's DMA
            asm volatile("s_wait_asynccnt 0x2" ::: "memory");  // tile kt landed
        } else {
            asm volatile("s_wait_asynccnt 0x0" ::: "memory");
        }
        __syncthreads();                           // all waves' DMA visible

        const int j0 = kt * 16;
        const unsigned w  = brow[j0 >> 5];
        const int      sh = j0 & 16;
        float pv[8];
#pragma unroll
        for (int kstep = 0; kstep < 4; ++kstep) {
#pragma unroll
            for (int t = 0; t < 2; ++t) {
                const int jl = 4 * kstep + 2 * half + t;
                float e = s1v + s2s[j0 + jl];
                e = (e < 0.f) ? 0.2f * e : e;
                const bool ok = (w >> (sh + jl)) & 1u;
                float p;
                if (uni)       p = 1.0f / (float)NN;
                else if (ok)   p = __expf(e - rmx) * rinv;
                else           p = 0.0f;
                pv[2 * kstep + t] = p;
            }
        }

#pragma unroll
        for (int c = 0; c < 4; ++c) {
            const int col = (chalf * 4 + c) * 16 + mloc;
#pragma unroll
            for (int kstep = 0; kstep < 4; ++kstep) {
                v2f A; A.x = pv[2 * kstep]; A.y = pv[2 * kstep + 1];
                const int kr = 4 * kstep + 2 * half;
                v2f B; B.x = hk[buf][kr][col]; B.y = hk[buf][kr + 1][col];
                acc[c] = __builtin_amdgcn_wmma_f32_16x16x4_f32(
                             false, A, false, B, (short)0, acc[c], false, false);
            }
        }
        __syncthreads();      // buf safe to overwrite at next iteration's issue
    }

    // Epilogue: transpose to (n, head*OUTF + o) with ELU.
#pragma unroll
    for (int c = 0; c < 4; ++c) {
        const int col = head * OUTF + (chalf * 4 + c) * 16 + mloc;
#pragma unroll
        for (int r = 0; r < 8; ++r) {
            const int n = m0 + r + 8 * half;
            float v = acc[c][r];
            v = (v > 0.f) ? v : (__expf(v) - 1.0f);
            out[(size_t)n * (NHEAD * OUTF) + col] = v;
        }
    }
}

// ---------------------------------------------------------------------------
extern "C" void kernel_launch(void* const* d_in, const int* in_sizes, int n_in,
                              void* d_out, int out_size, void* d_ws, size_t ws_size,
                              hipStream_t stream) {
    const float* x   = (const float*)d_in[0];   // (8192, 256)
    const int*   adj = (const int*)  d_in[1];   // (8192, 8192)
    const int*   idx = (const int*)  d_in[2];   // (4096,)
    const float* W   = (const float*)d_in[3];   // (4, 256, 128)
    const float* a   = (const float*)d_in[4];   // (4, 256)
    float*       out = (float*)d_out;           // (4096, 512)

    float*    h_ws = (float*)d_ws;                               // 8 MB
    float*    s1   = h_ws + (size_t)NHEAD * NN * OUTF;
    float*    s2   = s1 + (size_t)NHEAD * NN;
    float*    rmx  = s2 + (size_t)NHEAD * NN;
    float*    rsm  = rmx + (size_t)NHEAD * NN;
    unsigned* bits = (unsigned*)(rsm + (size_t)NHEAD * NN);      // 2 MB

    k_pack_adj<<<dim3(NN), dim3(128), 0, stream>>>(adj, idx, bits);
    k_gemm_h<<<dim3(NN / 16, NHEAD), dim3(256), 0, stream>>>(x, idx, W, h_ws);
    k_scores<<<dim3((NHEAD * NN) / 256), dim3(256), 0, stream>>>(h_ws, a, s1, s2);
    k_stats<<<dim3((NHEAD * NN) / 4), dim3(128), 0, stream>>>(s1, s2, bits, rmx, rsm);
    k_attn<<<dim3(NN / 64, NHEAD), dim3(256), 0, stream>>>(h_ws, s1, s2, bits,
                                                           rmx, rsm, out);
}